// FeatureExtractor_29506425323899
// MI455X (gfx1250) — compile-verified
//
#include <hip/hip_runtime.h>
#include <math.h>

typedef __attribute__((ext_vector_type(2))) float v2f;
typedef __attribute__((ext_vector_type(8))) float v8f;

#define NNODES 50000
#define NEDGES 800000
#define FIN 128
#define FOUT 64
#define KSEL 25000       // ceil(0.5 * N)
#define SORTM 65536      // next pow2 >= N

// ---------------------------------------------------------------- utilities

__global__ void zero4_kernel(float4* ptr, int count4) {
    int t = blockIdx.x * blockDim.x + threadIdx.x;
    if (t < count4) ptr[t] = make_float4(0.f, 0.f, 0.f, 0.f);
}

__device__ __forceinline__ void atomic_fadd(float* p, float v) {
    // lowers to global_atomic_add_f32 (L2-side RMW) on gfx1250
    __hip_atomic_fetch_add(p, v, __ATOMIC_RELAXED, __HIP_MEMORY_SCOPE_AGENT);
}

// ---------------------------------------------------------------- degree / dinv

__global__ void degree_kernel(const long long* __restrict__ src, float* __restrict__ deg, int E) {
    int e = blockIdx.x * blockDim.x + threadIdx.x;
    if (e < E) atomic_fadd(&deg[(int)src[e]], 1.0f);
}

__global__ void dinv_kernel(const float* __restrict__ deg, float* __restrict__ dinv, int n) {
    int i = blockIdx.x * blockDim.x + threadIdx.x;
    if (i < n) {
        float d = deg[i];
        dinv[i] = d > 0.f ? rsqrtf(d) : 0.f;
    }
}

// ---------------------------------------------------------------- edge propagation
// out[dst] += -(dinv[src]*dinv[dst]) * z[src]   (one thread per (edge, 4-feature group))

__global__ void prop_kernel(const long long* __restrict__ src, const long long* __restrict__ dst,
                            const float* __restrict__ dinv, const float* __restrict__ z,
                            float* __restrict__ out, int E) {
    long long t = (long long)blockIdx.x * blockDim.x + threadIdx.x;
    int e = (int)(t >> 5);        // 32 groups of 4 floats = 128 features
    int g = (int)(t & 31);
    if (e >= E) return;
    int s = (int)src[e];
    int d = (int)dst[e];
    float w = -(dinv[s] * dinv[d]);
    const float4 zv = *(const float4*)(z + (long long)s * FIN + g * 4);
    float* o = out + (long long)d * FIN + g * 4;
    atomic_fadd(o + 0, w * zv.x);
    atomic_fadd(o + 1, w * zv.y);
    atomic_fadd(o + 2, w * zv.z);
    atomic_fadd(o + 3, w * zv.w);
}

// ---------------------------------------------------------------- WMMA GEMM + bias + ELU
// h = ELU( x@W0 + Tx1@W1 + (2*P2 - x)@W2 + b )
// One wave per 16-row strip; 4 v8f accumulators cover all 64 output cols.

__global__ void __launch_bounds__(256)
cheb_gemm_kernel(const float* __restrict__ x, const float* __restrict__ tx1,
                 const float* __restrict__ p2, const float* __restrict__ W,
                 const float* __restrict__ bias, float* __restrict__ h) {
    int wave = (int)((blockIdx.x * blockDim.x + threadIdx.x) >> 5);
    int lane = threadIdx.x & 31;
    const int ntiles = NNODES / 16;          // 3125 exactly
    if (wave >= ntiles) return;
    int row0 = wave * 16;
    int hi   = lane >> 4;                    // 0: lanes 0-15, 1: lanes 16-31
    int lr   = lane & 15;
    long long arow = (long long)(row0 + lr) * FIN;

    v8f acc0 = {}, acc1 = {}, acc2 = {}, acc3 = {};

    for (int term = 0; term < 3; ++term) {
        const float* Wk = W + term * FIN * FOUT;
        for (int kk = 0; kk < FIN; kk += 4) {
            int ka = kk + (hi ? 2 : 0);      // A: VGPR0 holds K0/K2, VGPR1 holds K1/K3
            v2f a;
            if (term == 0) {
                a.x = x[arow + ka];
                a.y = x[arow + ka + 1];
            } else if (term == 1) {
                a.x = tx1[arow + ka];
                a.y = tx1[arow + ka + 1];
            } else {                          // T2 = 2*prop(T1) - T0, fused in-register
                a.x = 2.0f * p2[arow + ka]     - x[arow + ka];
                a.y = 2.0f * p2[arow + ka + 1] - x[arow + ka + 1];
            }
            // B: 4x16 tile, row K striped across lanes within a VGPR
            v2f b0, b1, b2, b3;
            b0.x = Wk[ka * FOUT +  0 + lr];  b0.y = Wk[(ka + 1) * FOUT +  0 + lr];
            b1.x = Wk[ka * FOUT + 16 + lr];  b1.y = Wk[(ka + 1) * FOUT + 16 + lr];
            b2.x = Wk[ka * FOUT + 32 + lr];  b2.y = Wk[(ka + 1) * FOUT + 32 + lr];
            b3.x = Wk[ka * FOUT + 48 + lr];  b3.y = Wk[(ka + 1) * FOUT + 48 + lr];
            acc0 = __builtin_amdgcn_wmma_f32_16x16x4_f32(false, a, false, b0, (short)0, acc0, false, false);
            acc1 = __builtin_amdgcn_wmma_f32_16x16x4_f32(false, a, false, b1, (short)0, acc1, false, false);
            acc2 = __builtin_amdgcn_wmma_f32_16x16x4_f32(false, a, false, b2, (short)0, acc2, false, false);
            acc3 = __builtin_amdgcn_wmma_f32_16x16x4_f32(false, a, false, b3, (short)0, acc3, false, false);
        }
    }

    // Epilogue: bias + ELU, D layout: VGPR r -> M=r (lanes 0-15) / M=r+8 (lanes 16-31), N=lr
    float bcol0 = bias[ 0 + lr], bcol1 = bias[16 + lr], bcol2 = bias[32 + lr], bcol3 = bias[48 + lr];
#pragma unroll
    for (int r = 0; r < 8; ++r) {
        int m = r + hi * 8;
        long long o = (long long)(row0 + m) * FOUT;
        float v;
        v = acc0[r] + bcol0; v = v > 0.f ? v : (expf(v) - 1.0f); h[o +  0 + lr] = v;
        v = acc1[r] + bcol1; v = v > 0.f ? v : (expf(v) - 1.0f); h[o + 16 + lr] = v;
        v = acc2[r] + bcol2; v = v > 0.f ? v : (expf(v) - 1.0f); h[o + 32 + lr] = v;
        v = acc3[r] + bcol3; v = v > 0.f ? v : (expf(v) - 1.0f); h[o + 48 + lr] = v;
    }
}

// ---------------------------------------------------------------- projection / score

__global__ void pnorm_kernel(const float* __restrict__ p, float* __restrict__ pn) {
    if (threadIdx.x == 0 && blockIdx.x == 0) {
        float s = 0.f;
        for (int i = 0; i < FOUT; ++i) s += p[i] * p[i];
        *pn = sqrtf(s);
    }
}

__global__ void score_kernel(const float* __restrict__ h, const float* __restrict__ p,
                             const float* __restrict__ pn, float* __restrict__ score, int n) {
    int i = blockIdx.x * blockDim.x + threadIdx.x;
    if (i >= n) return;
    float s = 0.f;
    const float* hr = h + (long long)i * FOUT;
#pragma unroll 8
    for (int f = 0; f < FOUT; ++f) s += hr[f] * p[f];
    score[i] = s / *pn;
}

// ---------------------------------------------------------------- top-k via bitonic sort
// key = ordered(score) << 32 | (~idx)  -> descending sort gives score desc, idx asc on ties

__device__ __forceinline__ unsigned int float_to_ordered(float f) {
    unsigned int u = __float_as_uint(f);
    return u ^ (((int)u >> 31) | 0x80000000u);
}

__global__ void pack_keys_kernel(const float* __restrict__ score, unsigned long long* __restrict__ keys) {
    int i = blockIdx.x * blockDim.x + threadIdx.x;
    if (i >= SORTM) return;
    if (i < NNODES) {
        keys[i] = ((unsigned long long)float_to_ordered(score[i]) << 32)
                | (unsigned long long)(0xFFFFFFFFu - (unsigned int)i);
    } else {
        keys[i] = 0ull;   // sorts to the tail in descending order
    }
}

__global__ void bitonic_step_kernel(unsigned long long* __restrict__ keys, unsigned int j, unsigned int k) {
    unsigned int i = blockIdx.x * blockDim.x + threadIdx.x;
    if (i >= SORTM) return;
    unsigned int ixj = i ^ j;
    if (ixj <= i) return;
    unsigned long long a = keys[i];
    unsigned long long b = keys[ixj];
    bool desc = ((i & k) == 0);              // descending overall
    bool swap = desc ? (a < b) : (a > b);
    if (swap) { keys[i] = b; keys[ixj] = a; }
}

// ---------------------------------------------------------------- gather + gate + emit idx

__global__ void gather_gate_kernel(const unsigned long long* __restrict__ keys,
                                   const float* __restrict__ h, const float* __restrict__ score,
                                   float* __restrict__ out, int* __restrict__ out_idx) {
    long long t = (long long)blockIdx.x * blockDim.x + threadIdx.x;
    int i = (int)(t >> 6);                   // selected slot
    int f = (int)(t & 63);                   // feature
    if (i >= KSEL) return;
    unsigned long long key = keys[i];
    unsigned int idx = 0xFFFFFFFFu - (unsigned int)(key & 0xFFFFFFFFull);
    float g = tanhf(score[idx]);
    out[(long long)i * FOUT + f] = h[(long long)idx * FOUT + f] * g;
    if (f == 0) out_idx[i] = (int)idx;
}

// ---------------------------------------------------------------- launch

extern "C" void kernel_launch(void* const* d_in, const int* in_sizes, int n_in,
                              void* d_out, int out_size, void* d_ws, size_t ws_size,
                              hipStream_t stream) {
    const float*     x     = (const float*)d_in[0];
    const long long* edges = (const long long*)d_in[1];   // int64 [2, E]
    const float*     W     = (const float*)d_in[2];       // [3,128,64]
    const float*     bias  = (const float*)d_in[3];       // [64]
    const float*     p     = (const float*)d_in[4];       // [64]
    const long long* esrc  = edges;
    const long long* edst  = edges + NEDGES;

    // workspace layout (256B aligned)
    char* ws = (char*)d_ws;
    size_t off = 0;
    auto take = [&](size_t bytes) { size_t r = off; off = (off + bytes + 255) & ~(size_t)255; return r; };
    float*              deg   = (float*)(ws + take((size_t)NNODES * 4));
    float*              dinv  = (float*)(ws + take((size_t)NNODES * 4));
    float*              tx1   = (float*)(ws + take((size_t)NNODES * FIN * 4));
    float*              pp2   = (float*)(ws + take((size_t)NNODES * FIN * 4));
    float*              h     = (float*)(ws + take((size_t)NNODES * FOUT * 4));
    float*              score = (float*)(ws + take((size_t)NNODES * 4));
    unsigned long long* keys  = (unsigned long long*)(ws + take((size_t)SORTM * 8));
    float*              pn    = (float*)(ws + take(256));

    float* out     = (float*)d_out;
    int*   out_idx = (int*)(out + (long long)KSEL * FOUT);

    const int B = 256;

    // zero the atomic accumulation buffers
    zero4_kernel<<<(NNODES / 4 + B - 1) / B, B, 0, stream>>>((float4*)deg, NNODES / 4);
    zero4_kernel<<<(NNODES * FIN / 4 + B - 1) / B, B, 0, stream>>>((float4*)tx1, NNODES * FIN / 4);
    zero4_kernel<<<(NNODES * FIN / 4 + B - 1) / B, B, 0, stream>>>((float4*)pp2, NNODES * FIN / 4);

    // symmetric normalization weights
    degree_kernel<<<(NEDGES + B - 1) / B, B, 0, stream>>>(esrc, deg, NEDGES);
    dinv_kernel<<<(NNODES + B - 1) / B, B, 0, stream>>>(deg, dinv, NNODES);

    // Tx1 = L_hat @ x ; P2 = L_hat @ Tx1
    long long pt = (long long)NEDGES * 32;
    prop_kernel<<<(unsigned)((pt + B - 1) / B), B, 0, stream>>>(esrc, edst, dinv, x, tx1, NEDGES);
    prop_kernel<<<(unsigned)((pt + B - 1) / B), B, 0, stream>>>(esrc, edst, dinv, tx1, pp2, NEDGES);

    // h = ELU(x@W0 + Tx1@W1 + (2*P2 - x)@W2 + b)   (WMMA f32)
    int nwaves = NNODES / 16;                       // 3125
    int nblk   = (nwaves * 32 + B - 1) / B;
    cheb_gemm_kernel<<<nblk, B, 0, stream>>>(x, tx1, pp2, W, bias, h);

    // score = (h @ p) / ||p||
    pnorm_kernel<<<1, 64, 0, stream>>>(p, pn);
    score_kernel<<<(NNODES + B - 1) / B, B, 0, stream>>>(h, p, pn, score, NNODES);

    // exact sorted top-k via bitonic sort on packed keys
    pack_keys_kernel<<<(SORTM + B - 1) / B, B, 0, stream>>>(score, keys);
    for (unsigned int k2 = 2; k2 <= SORTM; k2 <<= 1)
        for (unsigned int j = k2 >> 1; j > 0; j >>= 1)
            bitonic_step_kernel<<<(SORTM + B - 1) / B, B, 0, stream>>>(keys, j, k2);

    // gate selected rows and emit indices
    long long gt = (long long)KSEL * FOUT;
    gather_gate_kernel<<<(unsigned)((gt + B - 1) / B), B, 0, stream>>>(keys, h, score, out, out_idx);
}